// GRUAttenDecoder_25546465477082
// MI455X (gfx1250) — compile-verified
//
#include <hip/hip_runtime.h>
#include <hip/hip_bf16.h>

// Problem constants (match reference)
#define Bsz   32
#define Tlen  64
#define Slen  128
#define Hdim  512
#define ATTN  256
#define EMBD  256
#define VM1   31999   // VOCAB-1

typedef __attribute__((ext_vector_type(2))) float v2f;
typedef __attribute__((ext_vector_type(8))) float v8f;
typedef __attribute__((ext_vector_type(4))) int   v4i;

// D = A(16x4, f32) * B(4x16, f32) + C  -- native fp32 WMMA on CDNA5
__device__ __forceinline__ v8f wmma4(v2f a, v2f b, v8f c) {
  return __builtin_amdgcn_wmma_f32_16x16x4_f32(
      /*neg_a=*/false, a, /*neg_b=*/false, b,
      /*c_mod=*/(short)0, c, /*reuse_a=*/false, /*reuse_b=*/false);
}

// ---- CDNA5 async global->LDS copies (ASYNCcnt path), with safe fallback ----
#if defined(__has_builtin)
# if __has_builtin(__builtin_amdgcn_global_load_async_to_lds_b128) && \
     __has_builtin(__builtin_amdgcn_global_load_async_to_lds_b32)
#  define HAVE_ASYNC_LDS 1
# endif
#endif

typedef __attribute__((address_space(1))) v4i* gp128_t;
typedef __attribute__((address_space(3))) v4i* lp128_t;
typedef __attribute__((address_space(1))) int* gp32_t;
typedef __attribute__((address_space(3))) int* lp32_t;

__device__ __forceinline__ void async_copy_b128(const float* g, float* l) {
#ifdef HAVE_ASYNC_LDS
  __builtin_amdgcn_global_load_async_to_lds_b128((gp128_t)g, (lp128_t)l, 0, 0);
#else
  *(float4*)l = *(const float4*)g;
#endif
}
__device__ __forceinline__ void async_copy_b32(const float* g, float* l) {
#ifdef HAVE_ASYNC_LDS
  __builtin_amdgcn_global_load_async_to_lds_b32((gp32_t)g, (lp32_t)l, 0, 0);
#else
  *l = *g;
#endif
}
__device__ __forceinline__ void wait_async0() {
#ifdef HAVE_ASYNC_LDS
# if __has_builtin(__builtin_amdgcn_s_wait_asynccnt)
  __builtin_amdgcn_s_wait_asynccnt(0);
# else
  asm volatile("s_wait_asynccnt 0x0" ::: "memory");
# endif
#endif
}

// ---------------------------------------------------------------------------
// Kernel 1: enc_proj[B*S, ATTN] = encoder_outputs(B*S,512) @ W_enc(512,256)
// One wave per 16x16 output tile. grid = (16, 32), block=256.
// ---------------------------------------------------------------------------
__global__ void k_encproj(const float* __restrict__ A,
                          const float* __restrict__ Wenc,
                          float* __restrict__ D) {
  const int lane = threadIdx.x & 31, w = threadIdx.x >> 5;
  const int hl = lane >> 4, lr = lane & 15;
  const int row0 = (blockIdx.y * 8 + w) * 16;   // 0..4095 step 16
  const int col0 = blockIdx.x * 16;             // 0..255 step 16
  const float* Ap = A + (size_t)(row0 + lr) * 512 + 2 * hl;
  const float* Bp = Wenc + (size_t)(2 * hl) * 256 + col0 + lr;
  v8f c = {};
  for (int k = 0; k < 512; k += 4) {
    v2f a; a.x = Ap[k];                 a.y = Ap[k + 1];
    v2f b; b.x = Bp[(size_t)k * 256];   b.y = Bp[(size_t)(k + 1) * 256];
    c = wmma4(a, b, c);
  }
  float* Dp = D + col0 + lr;
#pragma unroll
  for (int i = 0; i < 8; ++i) Dp[(size_t)(row0 + i + 8 * hl) * 256] = c[i];
}

// ---------------------------------------------------------------------------
// Kernel 2: gxe[t*32+b, 1536] = embedding[y[b,t]] @ W_ih[:, :256]^T + b_ih
// Gather GEMM: M=2048, N=1536, K=256.  grid=(96,16), block=256 (8 waves).
// ---------------------------------------------------------------------------
__global__ void k_embgates(const int* __restrict__ y,
                           const float* __restrict__ emb,
                           const float* __restrict__ W_ih,
                           const float* __restrict__ b_ih,
                           float* __restrict__ gxe) {
  const int lane = threadIdx.x & 31, w = threadIdx.x >> 5;
  const int hl = lane >> 4, lr = lane & 15;
  const int rowtile = blockIdx.y * 8 + w;       // 0..127
  const int col0 = blockIdx.x * 16;             // 0..1535 step 16
  const int g = rowtile * 16 + lr;              // global row = t*32+b
  const int t = g >> 5, b = g & 31;
  const int eid = y[b * Tlen + t];
  const float* Ap = emb + (size_t)eid * EMBD + 2 * hl;
  const float* Bp = W_ih + (size_t)(col0 + lr) * 768 + 2 * hl;  // W_ih^T[k,n]
  v8f c = {};
  for (int k = 0; k < 256; k += 4) {
    v2f a;  a.x = Ap[k];  a.y = Ap[k + 1];
    v2f bb; bb.x = Bp[k]; bb.y = Bp[k + 1];
    c = wmma4(a, bb, c);
  }
  const float bias = b_ih[col0 + lr];
  float* Dp = gxe + (size_t)col0 + lr;
#pragma unroll
  for (int i = 0; i < 8; ++i)
    Dp[(size_t)(rowtile * 16 + i + 8 * hl) * 1536] = c[i] + bias;
}

// ---------------------------------------------------------------------------
// Kernel 3 (per step): attention. One block per batch row, 256 threads.
// ---------------------------------------------------------------------------
__global__ void k_attn(int t, const float* __restrict__ hp,
                       const float* __restrict__ enc_proj,
                       const float* __restrict__ enc_out,
                       const int* __restrict__ mask,
                       const float* __restrict__ W_dec,
                       const float* __restrict__ vvec,
                       float* __restrict__ ctx,
                       float* __restrict__ attn_out) {
  const int b = blockIdx.x, tid = threadIdx.x;
  __shared__ float sh[512];
  __shared__ float sdec[256];
  __shared__ float sE[128];
  __shared__ float sAW[128];
  sh[tid] = hp[b * 512 + tid];
  sh[tid + 256] = hp[b * 512 + 256 + tid];
  __syncthreads();
  float acc = 0.f;
  for (int k = 0; k < 512; ++k) acc += sh[k] * W_dec[k * 256 + tid];
  sdec[tid] = acc;
  __syncthreads();
  const int w = tid >> 5, lane = tid & 31;
  for (int i = 0; i < 16; ++i) {
    const int s = w * 16 + i;
    const float* ep = enc_proj + ((size_t)b * Slen + s) * 256;
    float e = 0.f;
    for (int j = lane; j < 256; j += 32) e += tanhf(ep[j] + sdec[j]) * vvec[j];
    for (int off = 16; off > 0; off >>= 1) e += __shfl_xor(e, off, 32);
    if (lane == 0) sE[s] = (mask[b * Slen + s] == 0) ? -1.0e9f : e;
  }
  __syncthreads();
  float m = -3.4e38f;
  for (int s = 0; s < 128; ++s) m = fmaxf(m, sE[s]);
  float denom = 0.f;
  for (int s = 0; s < 128; ++s) denom += __expf(sE[s] - m);
  if (tid < 128) {
    const float a = __expf(sE[tid] - m) / denom;
    sAW[tid] = a;
    attn_out[((size_t)b * Tlen + t) * Slen + tid] = a;
  }
  __syncthreads();
  float c0 = 0.f, c1 = 0.f;
  for (int s = 0; s < 128; ++s) {
    const float* eo = enc_out + ((size_t)b * Slen + s) * 512;
    const float a = sAW[s];
    c0 += a * eo[tid];
    c1 += a * eo[tid + 256];
  }
  ctx[b * 512 + tid] = c0;
  ctx[b * 512 + 256 + tid] = c1;
}

// ---------------------------------------------------------------------------
// Kernel 4 (per step): gate GEMMs via WMMA.
//   gA = gxe[t] + ctx @ W_ih[:,256:768]^T   (input-side gates, K=512)
//   gB = b_hh  + h_prev @ W_hh^T            (hidden-side gates, K=512)
// ---------------------------------------------------------------------------
__global__ void k_gates(int t, const float* __restrict__ hp,
                        const float* __restrict__ ctx,
                        const float* __restrict__ gxe,
                        const float* __restrict__ W_ih,
                        const float* __restrict__ W_hh,
                        const float* __restrict__ b_hh,
                        float* __restrict__ gA, float* __restrict__ gB) {
  const int lane = threadIdx.x & 31, w = threadIdx.x >> 5;
  const int hl = lane >> 4, lr = lane & 15;
  const int id = blockIdx.x * 8 + w;       // 0..191
  const int row0 = blockIdx.y * 16;        // 0 or 16
  const bool isA = id < 96;
  const int col0 = (isA ? id : id - 96) * 16;
  v8f c = {};
  if (isA) {
    const float* Gp = gxe + (size_t)(t * 32) * 1536 + col0 + lr;
#pragma unroll
    for (int i = 0; i < 8; ++i) c[i] = Gp[(size_t)(row0 + i + 8 * hl) * 1536];
    const float* Ap = ctx + (size_t)(row0 + lr) * 512 + 2 * hl;
    const float* Bp = W_ih + (size_t)(col0 + lr) * 768 + 256 + 2 * hl;
    for (int k = 0; k < 512; k += 4) {
      v2f a;  a.x = Ap[k];  a.y = Ap[k + 1];
      v2f bb; bb.x = Bp[k]; bb.y = Bp[k + 1];
      c = wmma4(a, bb, c);
    }
    float* Dp = gA + col0 + lr;
#pragma unroll
    for (int i = 0; i < 8; ++i) Dp[(size_t)(row0 + i + 8 * hl) * 1536] = c[i];
  } else {
    const float bias = b_hh[col0 + lr];
#pragma unroll
    for (int i = 0; i < 8; ++i) c[i] = bias;
    const float* Ap = hp + (size_t)(row0 + lr) * 512 + 2 * hl;
    const float* Bp = W_hh + (size_t)(col0 + lr) * 512 + 2 * hl;
    for (int k = 0; k < 512; k += 4) {
      v2f a;  a.x = Ap[k];  a.y = Ap[k + 1];
      v2f bb; bb.x = Bp[k]; bb.y = Bp[k + 1];
      c = wmma4(a, bb, c);
    }
    float* Dp = gB + col0 + lr;
#pragma unroll
    for (int i = 0; i < 8; ++i) Dp[(size_t)(row0 + i + 8 * hl) * 1536] = c[i];
  }
}

// ---------------------------------------------------------------------------
// Kernel 5 (per step): GRU elementwise update -> H_all[t].
// ---------------------------------------------------------------------------
__global__ void k_hupd(int t, const float* __restrict__ hp,
                       const float* __restrict__ gA,
                       const float* __restrict__ gB,
                       float* __restrict__ Hall) {
  const int idx = blockIdx.x * 256 + threadIdx.x;  // 0..16383
  const int b = idx >> 9, j = idx & 511;
  const float* a = gA + b * 1536;
  const float* g = gB + b * 1536;
  const float r = 1.f / (1.f + __expf(-(a[j] + g[j])));
  const float z = 1.f / (1.f + __expf(-(a[512 + j] + g[512 + j])));
  const float n = tanhf(a[1024 + j] + r * g[1024 + j]);
  const float h0 = hp[b * 512 + j];
  Hall[((size_t)t * 32 + b) * 512 + j] = (1.f - z) * n + z * h0;
}

// ---------------------------------------------------------------------------
// Kernel 6: out[b,t,:] = H_all[t,b,:] @ W_fc(512,31999) + b_fc
// Blocked WMMA GEMM, M=2048 N=31999 K=512.
//   block tile 64(M) x 128(N), K staged in chunks of 32 via async->LDS.
//   8 waves in 2x4 grid; each wave: 2x2 register block of 16x16 WMMA tiles.
// LDS strides padded (A:36, B:144 floats) -> bank-conflict-free WMMA reads,
// while A rows stay 16B-aligned for b128 async copies.
// grid = (ceil(31999/128)=250, 2048/64=32), block = 256.
// ---------------------------------------------------------------------------
__global__ void k_final(const float* __restrict__ Hall,
                        const float* __restrict__ W_fc,
                        const float* __restrict__ b_fc,
                        float* __restrict__ out) {
  __shared__ __align__(16) float As[64 * 36];    //  9,216 B
  __shared__ __align__(16) float Bsh[32 * 144];  // 18,432 B
  const int tid = threadIdx.x;
  const int lane = tid & 31, w = tid >> 5;
  const int hl = lane >> 4, lr = lane & 15;
  const int mw = w >> 2, nw = w & 3;             // wave grid 2x4
  const int m0 = blockIdx.y * 64;                // row block
  const int n0 = blockIdx.x * 128;               // col block
  __builtin_prefetch(Hall + (size_t)m0 * 512, 0, 3);  // global_prefetch_b8

  v8f c00 = {}, c01 = {}, c10 = {}, c11 = {};

  for (int k0 = 0; k0 < 512; k0 += 32) {
    // ---- stage A panel 64x32 (16B-aligned -> async b128) ----
#pragma unroll
    for (int j = 0; j < 2; ++j) {
      const int idx = tid + j * 256;             // 0..511 b128 transfers
      const int r = idx >> 3;                    // row 0..63
      const int kk = (idx & 7) * 4;              // 0..28
      async_copy_b128(Hall + (size_t)(m0 + r) * 512 + k0 + kk,
                      &As[r * 36 + kk]);
    }
    // ---- stage B panel 32x128 (odd row stride -> async b32) ----
#pragma unroll
    for (int it = 0; it < 16; ++it) {
      const int e = it * 256 + tid;              // 0..4095
      const int kk = e >> 7;                     // 0..31
      const int nn = e & 127;
      float* dst = &Bsh[kk * 144 + nn];
      if (n0 + nn < VM1)
        async_copy_b32(W_fc + (size_t)(k0 + kk) * VM1 + n0 + nn, dst);
      else
        *dst = 0.f;
    }
    wait_async0();
    __syncthreads();
    // ---- compute: 8 WMMA per K-step pair across the 2x2 register block ----
    for (int kk = 0; kk < 32; kk += 4) {
      const float* Ap = &As[(mw * 32 + lr) * 36 + kk + 2 * hl];
      v2f a0; a0.x = Ap[0];       a0.y = Ap[1];
      v2f a1; a1.x = Ap[16 * 36]; a1.y = Ap[16 * 36 + 1];
      const float* Bp = &Bsh[(kk + 2 * hl) * 144 + nw * 32 + lr];
      v2f b0; b0.x = Bp[0];  b0.y = Bp[144];
      v2f b1; b1.x = Bp[16]; b1.y = Bp[144 + 16];
      c00 = wmma4(a0, b0, c00);
      c01 = wmma4(a0, b1, c01);
      c10 = wmma4(a1, b0, c10);
      c11 = wmma4(a1, b1, c11);
    }
    __syncthreads();
  }

  // ---- store 4 subtiles: row g = t*32+b -> out[b,t,:] ----
  auto store_tile = [&](v8f c, int ms, int ns) {
    const int n = n0 + nw * 32 + ns * 16 + lr;
    if (n >= VM1) return;
    const float bias = b_fc[n];
#pragma unroll
    for (int i = 0; i < 8; ++i) {
      const int g = m0 + mw * 32 + ms * 16 + i + 8 * hl;
      const int t = g >> 5, b = g & 31;
      out[((size_t)b * Tlen + t) * VM1 + n] = c[i] + bias;
    }
  };
  store_tile(c00, 0, 0);
  store_tile(c01, 0, 1);
  store_tile(c10, 1, 0);
  store_tile(c11, 1, 1);
}

// ---------------------------------------------------------------------------
extern "C" void kernel_launch(void* const* d_in, const int* in_sizes, int n_in,
                              void* d_out, int out_size, void* d_ws, size_t ws_size,
                              hipStream_t stream) {
  (void)in_sizes; (void)n_in; (void)out_size; (void)ws_size;
  const int*   y        = (const int*)  d_in[0];
  const float* enc_out  = (const float*)d_in[1];
  const float* dec_init = (const float*)d_in[2];
  const int*   mask     = (const int*)  d_in[3];
  const float* emb      = (const float*)d_in[4];
  const float* W_enc    = (const float*)d_in[5];
  const float* W_dec    = (const float*)d_in[6];
  const float* vvec     = (const float*)d_in[7];
  const float* W_ih     = (const float*)d_in[8];
  const float* W_hh     = (const float*)d_in[9];
  const float* b_ih     = (const float*)d_in[10];
  const float* b_hh     = (const float*)d_in[11];
  const float* W_fc     = (const float*)d_in[12];
  const float* b_fc     = (const float*)d_in[13];

  float* outs = (float*)d_out;                        // (B,T,31999)
  float* attn = outs + (size_t)Bsz * Tlen * VM1;      // (B,T,128)

  float* ws       = (float*)d_ws;
  float* enc_proj = ws;                               // 4096*256
  float* gxe      = enc_proj + (size_t)4096 * 256;    // 2048*1536
  float* ctx      = gxe + (size_t)2048 * 1536;        // 32*512
  float* gA       = ctx + 32 * 512;                   // 32*1536
  float* gB       = gA + 32 * 1536;                   // 32*1536
  float* Hall     = gB + 32 * 1536;                   // 2048*512

  k_encproj<<<dim3(16, 32), 256, 0, stream>>>(enc_out, W_enc, enc_proj);
  k_embgates<<<dim3(96, 16), 256, 0, stream>>>(y, emb, W_ih, b_ih, gxe);

  for (int t = 0; t < Tlen; ++t) {
    const float* hp = (t == 0) ? dec_init : (Hall + (size_t)(t - 1) * 32 * 512);
    k_attn<<<32, 256, 0, stream>>>(t, hp, enc_proj, enc_out, mask, W_dec, vvec,
                                   ctx, attn);
    k_gates<<<dim3(24, 2), 256, 0, stream>>>(t, hp, ctx, gxe, W_ih, W_hh, b_hh,
                                             gA, gB);
    k_hupd<<<64, 256, 0, stream>>>(t, hp, gA, gB, Hall);
  }

  k_final<<<dim3(250, 32), 256, 0, stream>>>(Hall, W_fc, b_fc, outs);
}